// LinearKernel_7387343749284
// MI455X (gfx1250) — compile-verified
//
#include <hip/hip_runtime.h>

// Batched pairwise-dot (GEMM): out[b,n,m] = sum_d x1[b,n,d]*x2[b,m,d] + const
// B=4, N=M=4096, D=32, f32. Store-bandwidth bound (268 MB out @ 23.3 TB/s).
// Uses V_WMMA_F32_16X16X4_F32 (CDNA5 f32 matrix op), K-loop of 8, NT stores.

typedef __attribute__((ext_vector_type(2))) float v2f;
typedef __attribute__((ext_vector_type(8))) float v8f;

#define B_DIM 4
#define N_DIM 4096
#define M_DIM 4096
#define D_DIM 32

#define STRIP_W 64                      // each wave: 16 rows x 64 cols
#define MSTRIPS (M_DIM / STRIP_W)       // 64
#define NTILES  (N_DIM / 16)            // 256
#define WAVES_PER_BLOCK 8

__global__ __launch_bounds__(256) void bmm_wmma_f32_16x16x4(
    const float* __restrict__ x1,   // [B, N, D]
    const float* __restrict__ x2,   // [B, M, D]
    const float* __restrict__ cst,  // [1]
    float* __restrict__ out)        // [B, N, M]
{
    const int lane = threadIdx.x & 31;
    const int wave = threadIdx.x >> 5;

    // Flat strip id -> (batch, row-tile along N, 64-wide col strip along M)
    int s = blockIdx.x * WAVES_PER_BLOCK + wave;
    const int b = s / (MSTRIPS * NTILES);
    s -= b * (MSTRIPS * NTILES);
    const int trow = s / MSTRIPS;          // 16-row tile index (N dim)
    const int tcs  = s - trow * MSTRIPS;   // 64-col strip index (M dim)

    const int half = lane >> 4;            // 0: K=0,1  1: K=2,3 (per WMMA A/B layout)
    const int l    = lane & 15;            // row-within-tile (A) / col-within-tile (B)

    const float bias = cst[0];

    // ---- Load A once: x1 rows [trow*16 + l], full K=32 slice for this lane ----
    // 32-bit A-matrix 16x4 layout: VGPR0 = K=0 (lanes 0-15) / K=2 (lanes 16-31),
    // VGPR1 = K=1 / K=3 -> one b64 load per K-step lands in the right regs.
    const float* arow = x1 + ((size_t)b * N_DIM + (size_t)trow * 16 + l) * D_DIM + 2 * half;
    v2f a[8];
#pragma unroll
    for (int i = 0; i < 8; ++i)
        a[i] = *(const v2f*)(arow + 4 * i);

    const size_t out_row0 = ((size_t)b * N_DIM + (size_t)trow * 16) * (size_t)M_DIM;

    // ---- 4 adjacent 16x16 tiles along M, reusing A ----
#pragma unroll
    for (int j = 0; j < 4; ++j) {
        const int colbase = tcs * STRIP_W + j * 16;

        // B tile: 4x16 (KxN); B rows striped across lanes within a VGPR.
        // Lane l holds output-col (= x2 row) colbase+l; half selects K pair.
        const float* brow = x2 + ((size_t)b * M_DIM + (size_t)colbase + l) * D_DIM + 2 * half;
        v2f bb[8];
#pragma unroll
        for (int i = 0; i < 8; ++i)
            bb[i] = *(const v2f*)(brow + 4 * i);

        // Accumulator pre-loaded with the scalar bias (handles "+ const" free).
        v8f acc = {bias, bias, bias, bias, bias, bias, bias, bias};

#pragma unroll
        for (int i = 0; i < 8; ++i) {
            // 8 args: (neg_a, A, neg_b, B, c_mod, C, reuse_a, reuse_b)
            acc = __builtin_amdgcn_wmma_f32_16x16x4_f32(
                false, a[i], false, bb[i], (short)0, acc, false, false);
        }

        // ---- Store: C/D layout: VGPR r -> row r (lanes 0-15) / row r+8 (lanes 16-31),
        // col = lane&15. Non-temporal: output is write-once, keep L2 for x1/x2.
        const int col = colbase + l;
#pragma unroll
        for (int r = 0; r < 8; ++r) {
            const int row_in_tile = r + 8 * half;
            float* p = out + out_row0 + (size_t)row_in_tile * M_DIM + col;
            __builtin_nontemporal_store(acc[r], p);
        }
    }
}

extern "C" void kernel_launch(void* const* d_in, const int* in_sizes, int n_in,
                              void* d_out, int out_size, void* d_ws, size_t ws_size,
                              hipStream_t stream) {
    const float* x1  = (const float*)d_in[0];   // [4, 4096, 32]
    const float* x2  = (const float*)d_in[1];   // [4, 4096, 32]
    const float* cst = (const float*)d_in[2];   // [1]
    float* out = (float*)d_out;                 // [4, 4096, 4096]

    const int total_strips = B_DIM * NTILES * MSTRIPS;        // 65536
    const int blocks = total_strips / WAVES_PER_BLOCK;        // 8192

    bmm_wmma_f32_16x16x4<<<blocks, 256, 0, stream>>>(x1, x2, cst, out);
}